// Mask_10222022164974
// MI455X (gfx1250) — compile-verified
//
#include <hip/hip_runtime.h>
#include <stdint.h>

#define S_LEN           1024
#define TOPK            5
#define WAVES_PER_BLOCK 8
#define BLOCK_THREADS   (WAVES_PER_BLOCK * 32)

typedef float    __attribute__((ext_vector_type(4))) f32x4;
typedef int      __attribute__((ext_vector_type(4))) i32x4;

__global__ __launch_bounds__(BLOCK_THREADS)
void mask_topk_kernel(const int* __restrict__ x,
                      const float* __restrict__ inten,
                      float* __restrict__ out_mask_x,
                      float* __restrict__ out_token,
                      float* __restrict__ out_pos,
                      int B)
{
    __shared__ __align__(16) int sh_x[WAVES_PER_BLOCK * S_LEN];

    const int lane = threadIdx.x & 31;
    const int wave = threadIdx.x >> 5;
    const int row  = blockIdx.x * WAVES_PER_BLOCK + wave;
    if (row >= B) return;                      // wave-uniform branch

    const int*   xrow = x     + (size_t)row * S_LEN;
    const float* irow = inten + (size_t)row * S_LEN;
    int*         shx  = sh_x  + wave * S_LEN;

    // ---- 1) async DMA the x row into LDS; overlaps with top-k compute ----
    // VDST vgpr = LDS byte address (low 32 bits of the flat AS3 pointer),
    // VADDR pair = 64-bit global address. 8 issues x 32 lanes x 16B = 4KB row.
#pragma unroll
    for (int j = 0; j < 8; ++j) {
        int e = j * 128 + lane * 4;
        uint32_t lptr = (uint32_t)(uintptr_t)(shx + e);
        uint64_t gptr = (uint64_t)(uintptr_t)(xrow + e);
        asm volatile("global_load_async_to_lds_b128 %0, %1, off"
                     :: "v"(lptr), "v"(gptr) : "memory");
    }

    // hint prefetch of this wave's intensity stripe (global_prefetch_b8)
    __builtin_prefetch(irow + lane * 32, 0, 3);

    // ---- 2) load intensity row, 32 elems/lane, coalesced nontemporal b128 ----
    // Intensities are uniform[0,1) -> non-negative -> raw IEEE bits are an
    // order-preserving uint32 key. No sign transform needed.
    uint32_t k[32];
#pragma unroll
    for (int j = 0; j < 8; ++j) {
        f32x4 v = __builtin_nontemporal_load(
            (const f32x4*)(irow + j * 128 + lane * 4));
        k[j * 4 + 0] = __float_as_uint(v.x);
        k[j * 4 + 1] = __float_as_uint(v.y);
        k[j * 4 + 2] = __float_as_uint(v.z);
        k[j * 4 + 3] = __float_as_uint(v.w);
    }

    // ---- 3) five argmax passes; packed (key<<32)|~idx so ties pick low idx ----
    uint32_t used = 0;
    int top[TOPK];
#pragma unroll
    for (int p = 0; p < TOPK; ++p) {
        unsigned long long best = 0ull;
#pragma unroll
        for (int s = 0; s < 32; ++s) {
            int g = ((s >> 2) << 7) + (lane << 2) + (s & 3);
            uint32_t key = ((used >> s) & 1u) ? 0u : k[s];
            unsigned long long cand =
                ((unsigned long long)key << 32) | (uint32_t)(~(uint32_t)g);
            best = (cand > best) ? cand : best;
        }
#pragma unroll
        for (int off = 16; off >= 1; off >>= 1) {
            unsigned long long o = __shfl_xor(best, off, 32);
            best = (o > best) ? o : best;
        }
        int g = (int)(~(uint32_t)best) & (S_LEN - 1);
        top[p] = g;
        if (((g >> 2) & 31) == lane)            // owning lane retires the slot
            used |= (1u << (((g >> 7) << 2) | (g & 3)));
    }

    // ---- 4) deterministic 2-of-5 selection (stand-in for jax PRNG) ----
    uint32_t h = (uint32_t)row * 2654435761u;
    h ^= h >> 16; h *= 0x85ebca6bu; h ^= h >> 13;
    int a = (int)(h % 5u);
    int b = (a + 1 + (int)((h / 5u) & 3u)) % 5;
    int pos0 = top[0], pos1 = top[1];
#pragma unroll
    for (int i = 0; i < TOPK; ++i) {
        pos0 = (i == a) ? top[i] : pos0;
        pos1 = (i == b) ? top[i] : pos1;
    }

    // ---- 5) consume the async copy: masked streaming copy x -> out ----
    asm volatile("s_wait_asynccnt 0" ::: "memory");

    float* orow = out_mask_x + (size_t)row * S_LEN;
#pragma unroll
    for (int j = 0; j < 8; ++j) {
        int e = j * 128 + lane * 4;
        i32x4 t = *(const i32x4*)(shx + e);     // ds_load_b128 from LDS
        f32x4 o;
        o.x = (e + 0 == pos0 || e + 0 == pos1) ? 1.0f : (float)t.x;
        o.y = (e + 1 == pos0 || e + 1 == pos1) ? 1.0f : (float)t.y;
        o.z = (e + 2 == pos0 || e + 2 == pos1) ? 1.0f : (float)t.z;
        o.w = (e + 3 == pos0 || e + 3 == pos1) ? 1.0f : (float)t.w;
        __builtin_nontemporal_store(o, (f32x4*)(orow + e));  // b128 th:NT
    }

    // ---- 6) tokens + positions (tokens read from the LDS copy of x) ----
    if (lane == 0) {
        float* tkp = out_token + (size_t)row * TOPK;
        float* psp = out_pos   + (size_t)row * TOPK;
        f32x4 tk = { (float)shx[pos0], (float)shx[pos1], 0.0f, 0.0f };
        f32x4 ps = { (float)pos0,      (float)pos1,      0.0f, 0.0f };
        __builtin_nontemporal_store(tk, (f32x4*)tkp);
        __builtin_nontemporal_store(ps, (f32x4*)psp);
        tkp[4] = 0.0f;
        psp[4] = 0.0f;
    }
}

extern "C" void kernel_launch(void* const* d_in, const int* in_sizes, int n_in,
                              void* d_out, int out_size, void* d_ws, size_t ws_size,
                              hipStream_t stream) {
    (void)n_in; (void)d_ws; (void)ws_size; (void)out_size;
    const int*   x     = (const int*)d_in[0];     // [B, S] token ids
    const float* inten = (const float*)d_in[1];   // [B, 1, S] intensities
    // d_in[2] = max_pred (5), matches TOPK padding; hardcoded.

    const int B = in_sizes[0] / S_LEN;

    float* out        = (float*)d_out;
    float* out_mask_x = out;                                        // B*S
    float* out_token  = out + (size_t)B * S_LEN;                    // B*5
    float* out_pos    = out + (size_t)B * S_LEN + (size_t)B * TOPK; // B*5

    const int grid = (B + WAVES_PER_BLOCK - 1) / WAVES_PER_BLOCK;
    mask_topk_kernel<<<grid, BLOCK_THREADS, 0, stream>>>(
        x, inten, out_mask_x, out_token, out_pos, B);
}